// GlobalPointer_30777735643781
// MI455X (gfx1250) — compile-verified
//
#include <hip/hip_runtime.h>
#include <math.h>

// Problem constants (match reference)
#define B_      16
#define S_      512
#define DIM_    1024
#define NT_     9
#define H_      64
#define NCOL_   (NT_ * 2 * H_)   // 1152
#define MROWS_  (B_ * S_)        // 8192
#define MASKNEG 1000000000000.0f

typedef __attribute__((ext_vector_type(2))) float v2f;
typedef __attribute__((ext_vector_type(4))) float v4f;
typedef __attribute__((ext_vector_type(8))) float v8f;
typedef __attribute__((__vector_size__(16))) int vi4;   // for async builtin

// ---- CDNA5 async global->LDS path (guarded; falls back to VGPR staging) ----
#ifndef __has_builtin
#define __has_builtin(x) 0
#endif
#if __has_builtin(__builtin_amdgcn_global_load_async_to_lds_b128)
#define ASYNC_LDS 1
#else
#define ASYNC_LDS 0
#endif

#define AS1 __attribute__((address_space(1)))
#define AS3 __attribute__((address_space(3)))

__device__ __forceinline__ void copy16_to_lds(const float* __restrict__ g,
                                              float* __restrict__ l)
{
#if ASYNC_LDS
    // async DMA: 16B per lane, tracked by ASYNCcnt, no VGPR staging
    __builtin_amdgcn_global_load_async_to_lds_b128(
        (AS1 vi4*)g, (AS3 vi4*)l, 0, 0);
#else
    v4f v = *(const v4f*)g;
    l[0] = v.x; l[1] = v.y; l[2] = v.z; l[3] = v.w;
#endif
}

__device__ __forceinline__ void async_wait_all()
{
#if ASYNC_LDS
#if __has_builtin(__builtin_amdgcn_s_wait_asynccnt)
    __builtin_amdgcn_s_wait_asynccnt(0);
#else
    asm volatile("s_wait_asynccnt 0" ::: "memory");
#endif
#endif
}

// -------------------------------------------------------------------------
// Kernel 1: out = h @ W + b, fused bias + RoPE, scatter into Q/K workspace
// laid out as (b, head, s, 64) fp32 for contiguous stage-2 reads.
// 64x64 tile / block, 8 waves, 2x 16x16 subtiles per wave, K chunk 32,
// double-buffered async global->LDS staging.
// -------------------------------------------------------------------------
#define K1_MT 64
#define K1_NT 64
#define K1_KC 32
#define K1_AS (K1_KC + 4)    // 36 dwords: bank-conflict-free A rows
#define K1_BS (K1_NT + 16)   // 80 dwords: halves of wave hit disjoint banks

__global__ __launch_bounds__(256)
void gp_proj_rope_kernel(const float* __restrict__ hid,
                         const float* __restrict__ W,
                         const float* __restrict__ bias,
                         float* __restrict__ q_ws,
                         float* __restrict__ k_ws)
{
    __shared__ float As[2][K1_MT * K1_AS];
    __shared__ float Bs[2][K1_KC * K1_BS];

    const int tid  = threadIdx.x;
    const int lane = tid & 31;
    const int wid  = tid >> 5;
    const int l16  = lane & 15;
    const int half = lane >> 4;          // EXEC half: lanes 16-31 hold K+2

    const int n0 = blockIdx.x * K1_NT;
    const int m0 = blockIdx.y * K1_MT;

    const int sub_m  = wid >> 1;          // 0..3
    const int sub_nb = (wid & 1) * 2;     // 0 or 2

    // staging (2 float4 of A + 2 float4 of B per thread per chunk)
    auto stage = [&](int buf, int k0) {
        #pragma unroll
        for (int i = 0; i < 2; ++i) {
            int f  = tid + 256 * i;          // float4 index 0..511
            int r  = f >> 3;                 // 8 float4 per 32-dword A row
            int c4 = f & 7;
            copy16_to_lds(hid + (size_t)(m0 + r) * DIM_ + k0 + c4 * 4,
                          &As[buf][r * K1_AS + c4 * 4]);
        }
        #pragma unroll
        for (int i = 0; i < 2; ++i) {
            int f  = tid + 256 * i;
            int r  = f >> 4;                 // 16 float4 per 64-dword B row
            int c4 = f & 15;
            copy16_to_lds(W + (size_t)(k0 + r) * NCOL_ + n0 + c4 * 4,
                          &Bs[buf][r * K1_BS + c4 * 4]);
        }
    };

    v8f c0 = {}; v8f c1 = {};

    stage(0, 0);
    int buf = 0;
    for (int k0 = 0; k0 < DIM_; k0 += K1_KC) {
        async_wait_all();        // my async stores to LDS (cur buf) complete
        __syncthreads();         // everyone's staging visible; prev buf free
        if (k0 + K1_KC < DIM_)
            stage(buf ^ 1, k0 + K1_KC);   // overlap next fetch with compute

        const float* arow = &As[buf][(sub_m * 16 + l16) * K1_AS + 2 * half];
        const float* bbuf = &Bs[buf][0];
        #pragma unroll
        for (int kk = 0; kk < K1_KC; kk += 4) {
            v2f a = *(const v2f*)(arow + kk);         // A[m][kk+2h .. +1]
            int krow = kk + 2 * half;
            {
                int nl = (sub_nb + 0) * 16 + l16;
                v2f bv; bv.x = bbuf[krow * K1_BS + nl];
                        bv.y = bbuf[(krow + 1) * K1_BS + nl];
                c0 = __builtin_amdgcn_wmma_f32_16x16x4_f32(
                        false, a, false, bv, (short)0, c0, false, false);
            }
            {
                int nl = (sub_nb + 1) * 16 + l16;
                v2f bv; bv.x = bbuf[krow * K1_BS + nl];
                        bv.y = bbuf[(krow + 1) * K1_BS + nl];
                c1 = __builtin_amdgcn_wmma_f32_16x16x4_f32(
                        false, a, false, bv, (short)0, c1, false, false);
            }
        }
        buf ^= 1;
    }

    // Epilogue: bias + RoPE, write Q/K workspace (b, head, s, d)
    const float NEG_L2_1E4_OVER_32 = -13.287712379549449f / 32.0f; // -log2(1e4)/32
    #pragma unroll
    for (int j = 0; j < 2; ++j) {
        v8f c = j ? c1 : c0;
        int col  = n0 + (sub_nb + j) * 16 + l16;   // 0..1151
        int t    = col >> 7;                       // head
        int dd   = col & 127;
        int d    = dd & 63;
        bool isk = dd >= 64;
        float bv  = bias[col];
        float inv = exp2f((float)(d >> 1) * NEG_L2_1E4_OVER_32);
        float sgn = (d & 1) ? 1.0f : -1.0f;        // rotate sign
        float* dst = isk ? k_ws : q_ws;
        #pragma unroll
        for (int i = 0; i < 8; ++i) {
            float x = c[i] + bv;
            float partner = __shfl_xor(x, 1, 32);  // paired RoPE element
            int row = m0 + sub_m * 16 + i + 8 * half;
            int bb  = row >> 9;
            int s   = row & 511;
            float sv, cv;
            __sincosf((float)s * inv, &sv, &cv);
            float y = x * cv + sgn * partner * sv;
            dst[(((size_t)bb * NT_ + t) * S_ + s) * H_ + d] = y;
        }
    }
}

// -------------------------------------------------------------------------
// Kernel 2: logits[b,h,m,n] = (Q.Kt * pad - (1-pad)*1e12) / 8
// 64x64 logits tile / block; Q,K tiles staged async into LDS; K-dim = 64.
// -------------------------------------------------------------------------
#define K2_T  64
#define K2_LS (H_ + 4)   // 68 dwords: conflict-free for both EXEC halves

__global__ __launch_bounds__(256)
void gp_attn_kernel(const float* __restrict__ q_ws,
                    const float* __restrict__ k_ws,
                    const float* __restrict__ mask,
                    float* __restrict__ out)
{
    __shared__ float Qs[K2_T * K2_LS];
    __shared__ float Ks[K2_T * K2_LS];

    const int tid  = threadIdx.x;
    const int lane = tid & 31;
    const int wid  = tid >> 5;
    const int l16  = lane & 15;
    const int half = lane >> 4;

    const int z  = blockIdx.z;            // b*9 + head
    const int bb = z / NT_;
    const int n0 = blockIdx.x * K2_T;
    const int m0 = blockIdx.y * K2_T;

    const float* qbase = q_ws + (size_t)z * S_ * H_;
    const float* kbase = k_ws + (size_t)z * S_ * H_;

    // stage Q rows m0.. and K rows n0.. (64x64 f32 each; 4 float4 per thread)
    #pragma unroll
    for (int i = 0; i < 4; ++i) {
        int f  = tid + 256 * i;           // float4 index 0..1023
        int r  = f >> 4;                  // 16 float4 per row
        int c4 = f & 15;
        copy16_to_lds(qbase + (size_t)(m0 + r) * H_ + c4 * 4,
                      &Qs[r * K2_LS + c4 * 4]);
        copy16_to_lds(kbase + (size_t)(n0 + r) * H_ + c4 * 4,
                      &Ks[r * K2_LS + c4 * 4]);
    }
    async_wait_all();
    __syncthreads();

    const int sub_m  = wid >> 1;
    const int sub_nb = (wid & 1) * 2;

    v8f c0 = {}; v8f c1 = {};
    const float* qrow  = &Qs[(sub_m * 16 + l16) * K2_LS + 2 * half];
    const float* krow0 = &Ks[((sub_nb + 0) * 16 + l16) * K2_LS + 2 * half];
    const float* krow1 = &Ks[((sub_nb + 1) * 16 + l16) * K2_LS + 2 * half];
    #pragma unroll
    for (int kk = 0; kk < H_; kk += 4) {
        v2f a  = *(const v2f*)(qrow  + kk);
        v2f b0 = *(const v2f*)(krow0 + kk);   // B[kk][n] = K[n][kk]
        v2f b1 = *(const v2f*)(krow1 + kk);
        c0 = __builtin_amdgcn_wmma_f32_16x16x4_f32(
                false, a, false, b0, (short)0, c0, false, false);
        c1 = __builtin_amdgcn_wmma_f32_16x16x4_f32(
                false, a, false, b1, (short)0, c1, false, false);
    }

    const float scale = 0.125f;           // 1/sqrt(64)
    float* obase = out + (size_t)z * S_ * S_;
    #pragma unroll
    for (int j = 0; j < 2; ++j) {
        v8f c = j ? c1 : c0;
        int n = n0 + (sub_nb + j) * 16 + l16;
        float pad  = mask[bb * S_ + n];
        float moff = (1.0f - pad) * MASKNEG;
        #pragma unroll
        for (int i = 0; i < 8; ++i) {
            int m = m0 + sub_m * 16 + i + 8 * half;
            obase[(size_t)m * S_ + n] = (c[i] * pad - moff) * scale;
        }
    }
}

// -------------------------------------------------------------------------
extern "C" void kernel_launch(void* const* d_in, const int* in_sizes, int n_in,
                              void* d_out, int out_size, void* d_ws, size_t ws_size,
                              hipStream_t stream)
{
    const float* hid  = (const float*)d_in[0];   // (16,512,1024)
    const float* mask = (const float*)d_in[1];   // (16,512)
    const float* W    = (const float*)d_in[2];   // (1024,1152)
    const float* bias = (const float*)d_in[3];   // (1152,)

    float* q_ws = (float*)d_ws;
    float* k_ws = q_ws + (size_t)B_ * NT_ * S_ * H_;   // 18.9 MB each

    dim3 g1(NCOL_ / K1_NT, MROWS_ / K1_MT);  // (18, 128)
    gp_proj_rope_kernel<<<g1, 256, 0, stream>>>(hid, W, bias, q_ws, k_ws);

    dim3 g2(S_ / K2_T, S_ / K2_T, B_ * NT_); // (8, 8, 144)
    gp_attn_kernel<<<g2, 256, 0, stream>>>(q_ws, k_ws, mask, (float*)d_out);
}